// Codec_60069412602064
// MI455X (gfx1250) — compile-verified
//
#include <hip/hip_runtime.h>
#include <math.h>

#define B_IMG 16
#define CH 3
#define HH 512
#define WW 512
#define NOUT0 12          // 3*NPRED
#define K_TAPS 120        // 3 * (4*9 + 4)
#define LDS_W 520         // WW + 2*4
#define NBINS 256

typedef float v2f __attribute__((ext_vector_type(2)));
typedef float v8f __attribute__((ext_vector_type(8)));

__device__ __forceinline__ float leaky01(float v) { return v > 0.f ? v : 0.01f * v; }

__global__ __launch_bounds__(256) void codec_main(
    const float* __restrict__ x,
    const float* __restrict__ w0T, const float* __restrict__ w0L, const float* __restrict__ b0L,
    const float* __restrict__ w1,  const float* __restrict__ b1,
    const float* __restrict__ w2,  const float* __restrict__ b2,
    unsigned* __restrict__ gHistX, unsigned* __restrict__ gHistD,
    float* __restrict__ partD2, float* __restrict__ partX2)
{
    __shared__ float    sYuv[CH * 5 * LDS_W];   // 31200 B : rows h-4..h, padded cols
    __shared__ float    sWmat[K_TAPS * 16];     // 7680 B  : B matrix, N padded to 16
    __shared__ int      sTapOff[128];           // tap -> LDS offset LUT
    __shared__ float    sOut0[8][16][16];       // per-wave transpose scratch
    __shared__ unsigned sHX[CH * NBINS];
    __shared__ unsigned sHD[CH * NBINS];
    __shared__ float    sB0[16], sW1[NOUT0 * 4], sB1[NOUT0], sW2[CH * 4], sB2[CH];
    __shared__ float    sRed[256];

    const int tid = threadIdx.x;
    const int wg  = blockIdx.x;        // 0 .. 16*512-1
    const int n   = wg / HH;
    const int h   = wg % HH;

    // ---- stage layer-0 weight matrix (tap-major, N padded to 16) ----
    for (int i = tid; i < K_TAPS * 16; i += 256) {
        int k = i >> 4, c = i & 15;
        float v = 0.f;
        if (c < NOUT0) {
            int ic = k / 40, tt = k - 40 * ic;
            if (tt < 36) { int kh = tt / 9, kw = tt - 9 * kh; v = w0T[((c * 3 + ic) * 4 + kh) * 9 + kw]; }
            else         { v = w0L[(c * 3 + ic) * 4 + (tt - 36)]; }
        }
        sWmat[i] = v;
    }
    // ---- tap -> LDS address LUT (relative to pixel column w) ----
    if (tid < K_TAPS) {
        int k = tid;
        int ic = k / 40, tt = k - 40 * ic;
        int row, colAdd;
        if (tt < 36) { row = tt / 9; colAdd = tt - 9 * row; }  // top ctx: rows h-4..h-1, cols w-4..w+4
        else         { row = 4;      colAdd = tt - 36; }       // left ctx: row h, cols w-4..w-1
        sTapOff[k] = ic * (5 * LDS_W) + row * LDS_W + colAdd;  // add (w) at use site; LDS col = gw+4
    }
    if (tid < 16)        sB0[tid] = (tid < NOUT0) ? b0L[tid] : 0.f;
    if (tid < NOUT0 * 4) sW1[tid] = w1[tid];
    if (tid < NOUT0)     sB1[tid] = b1[tid];
    if (tid < CH * 4)    sW2[tid] = w2[tid];
    if (tid < CH)        sB2[tid] = b2[tid];
    for (int i = tid; i < CH * NBINS; i += 256) { sHX[i] = 0u; sHD[i] = 0u; }

    // ---- stage yuv rows h-4..h with zero padding ----
    const float* xr = x + (size_t)(n * 3 + 0) * HH * WW;
    const float* xg = x + (size_t)(n * 3 + 1) * HH * WW;
    const float* xb = x + (size_t)(n * 3 + 2) * HH * WW;
    for (int i = tid; i < 5 * LDS_W; i += 256) {
        int r = i / LDS_W, c = i - r * LDS_W;
        int gr = h - 4 + r, gc = c - 4;
        float rv = 0.f, gv = 0.f, bv = 0.f;
        if (gr >= 0 && gc >= 0 && gc < WW) {
            size_t o = (size_t)gr * WW + gc;
            rv = xr[o]; gv = xg[o]; bv = xb[o];
        }
        sYuv[0 * 5 * LDS_W + i] = rv - gv;   // Cr = r - g
        sYuv[1 * 5 * LDS_W + i] = gv;        // Y  = g
        sYuv[2 * 5 * LDS_W + i] = bv - gv;   // Cb = b - g
    }
    __syncthreads();

    const int lane = tid & 31;
    const int wave = tid >> 5;
    const int hi   = lane >> 4;   // lane half: selects K pair (A) / K pair (B)
    const int mn   = lane & 15;   // M index for A, N index for B/D

    float accD2 = 0.f, accX2 = 0.f;

    for (int t = 0; t < 4; ++t) {
        const int wbase = (wave * 4 + t) * 16;   // first pixel column of this 16-wide tile
        v8f acc = {0.f, 0.f, 0.f, 0.f, 0.f, 0.f, 0.f, 0.f};

        // ---- layer 0 as GEMM: D(16px x 16ch) = A(16x120 taps) x B(120x16), K split in 30 steps ----
        #pragma unroll
        for (int s = 0; s < 30; ++s) {
            int k  = 4 * s + 2 * hi;
            int o0 = sTapOff[k], o1 = sTapOff[k + 1];
            v2f a, b;
            a.x = sYuv[o0 + wbase + mn];
            a.y = sYuv[o1 + wbase + mn];
            b.x = sWmat[k * 16 + mn];
            b.y = sWmat[(k + 1) * 16 + mn];
            acc = __builtin_amdgcn_wmma_f32_16x16x4_f32(false, a, false, b, (short)0, acc,
                                                        false, false);
        }

        // ---- bias + leaky-relu, transpose into per-wave LDS scratch ----
        {
            float bias = sB0[mn];
            #pragma unroll
            for (int j = 0; j < 8; ++j) {
                int m = hi * 8 + j;                 // C/D layout: lanes 0-15 M=j, lanes 16-31 M=j+8
                sOut0[wave][m][mn] = leaky01(acc[j] + bias);
            }
        }

        // ---- per-pixel tail (grouped 1x1 layers, median clamp, stats) on lanes 0..15 ----
        if (lane < 16) {
            const int w_ = wbase + lane;
            float o0v[12];
            #pragma unroll
            for (int c = 0; c < 12; ++c) o0v[c] = sOut0[wave][lane][c];

            float o1v[12];
            #pragma unroll
            for (int oc = 0; oc < 12; ++oc) {
                int g4 = (oc >> 2) * 4;
                float s = sB1[oc];
                #pragma unroll
                for (int k = 0; k < 4; ++k) s += o0v[g4 + k] * sW1[oc * 4 + k];
                o1v[oc] = leaky01(s);
            }

            float g = sYuv[1 * 5 * LDS_W + 4 * LDS_W + (w_ + 4)];  // exact x[n,1,h,w]
            #pragma unroll
            for (int c = 0; c < CH; ++c) {
                float p = sB2[c];
                #pragma unroll
                for (int k = 0; k < 4; ++k) p += o1v[c * 4 + k] * sW2[c * 4 + k];
                float north = sYuv[c * 5 * LDS_W + 3 * LDS_W + (w_ + 4)];
                float west  = sYuv[c * 5 * LDS_W + 4 * LDS_W + (w_ + 3)];
                float ne    = sYuv[c * 5 * LDS_W + 3 * LDS_W + (w_ + 5)];
                float vmin = fminf(west, fminf(north, ne));
                float vmax = fmaxf(west, fmaxf(north, ne));
                // median3(p, vmin, vmax)
                float med = fmaxf(fminf(p, fmaxf(vmin, vmax)), fminf(vmin, vmax));
                float predv = med + ((c == 1) ? 0.f : g);
                predv = fminf(fmaxf(predv, -1.f), 1.f);
                float xv = (c == 1) ? g : x[((size_t)(n * 3 + c) * HH + h) * WW + w_];
                float d = xv - predv;
                accD2 += d * d;
                accX2 += xv * xv;

                if (xv >= -1.f && xv <= 1.f) {
                    int idx = (int)floorf((xv + 1.f) * 128.f);
                    idx = min(max(idx, 0), 255);
                    atomicAdd(&sHX[c * NBINS + idx], 1u);
                }
                float wd = fmodf(d + 1.f, 2.f) - 1.f;
                if (wd >= -1.f && wd <= 1.f) {
                    int idx = (int)floorf((wd + 1.f) * 128.f);
                    idx = min(max(idx, 0), 255);
                    atomicAdd(&sHD[c * NBINS + idx], 1u);
                }
            }
        }
    }

    // ---- deterministic per-block reduction of squared sums ----
    sRed[tid] = accD2;
    __syncthreads();
    for (int off = 128; off > 0; off >>= 1) {
        if (tid < off) sRed[tid] += sRed[tid + off];
        __syncthreads();
    }
    if (tid == 0) partD2[wg] = sRed[0];
    __syncthreads();
    sRed[tid] = accX2;
    __syncthreads();
    for (int off = 128; off > 0; off >>= 1) {
        if (tid < off) sRed[tid] += sRed[tid + off];
        __syncthreads();
    }
    if (tid == 0) partX2[wg] = sRed[0];
    __syncthreads();

    // ---- flush histograms (integer atomics: order-independent, deterministic) ----
    for (int i = tid; i < CH * NBINS; i += 256) {
        int c = i >> 8, bin = i & 255;
        unsigned vx = sHX[i], vd = sHD[i];
        if (vx) atomicAdd(&gHistX[(n * 3 + c) * NBINS + bin], vx);
        if (vd) atomicAdd(&gHistD[(n * 3 + c) * NBINS + bin], vd);
    }
}

__global__ __launch_bounds__(256) void codec_final(
    const unsigned* __restrict__ gHistX, const unsigned* __restrict__ gHistD,
    const float* __restrict__ partD2, const float* __restrict__ partX2,
    float* __restrict__ out)
{
    __shared__ float sRed[256];
    const int tid = threadIdx.x;
    const float RES = (float)(HH * WW);  // 262144 px per channel

    float sd = 0.f, sx = 0.f;
    for (int i = tid; i < B_IMG * HH; i += 256) { sd += partD2[i]; sx += partX2[i]; }

    float ex = 0.f, ed = 0.f;
    for (int i = tid; i < B_IMG * CH * NBINS; i += 256) {
        float px = (float)gHistX[i] / RES;
        if (px > 0.f) ex -= px * log2f(px);
        float pd = (float)gHistD[i] / RES;
        if (pd > 0.f) ed -= pd * log2f(pd);
    }

    float vals[4] = { sd, sx, ex, ed };
    float tot[4];
    for (int v = 0; v < 4; ++v) {
        __syncthreads();
        sRed[tid] = vals[v];
        __syncthreads();
        for (int off = 128; off > 0; off >>= 1) {
            if (tid < off) sRed[tid] += sRed[tid + off];
            __syncthreads();
        }
        tot[v] = sRed[0];
    }
    if (tid == 0) {
        const float NTOT = (float)((size_t)B_IMG * CH * HH * WW);  // 12582912
        out[0] = 255.0f * sqrtf(tot[0] / NTOT);   // loss1 (rmse of delta)
        out[1] = 255.0f * sqrtf(tot[1] / NTOT);   // loss0 (rmse of x)
        out[2] = tot[2] / (8.0f * (float)(B_IMG * CH));  // invCR0
        out[3] = tot[3] / (8.0f * (float)(B_IMG * CH));  // invCR1
    }
}

extern "C" void kernel_launch(void* const* d_in, const int* in_sizes, int n_in,
                              void* d_out, int out_size, void* d_ws, size_t ws_size,
                              hipStream_t stream)
{
    (void)in_sizes; (void)n_in; (void)out_size; (void)ws_size;
    const float* x   = (const float*)d_in[0];
    const float* w0T = (const float*)d_in[1];
    const float* w0L = (const float*)d_in[2];
    const float* b0L = (const float*)d_in[3];
    const float* w1  = (const float*)d_in[4];
    const float* b1  = (const float*)d_in[5];
    const float* w2  = (const float*)d_in[6];
    const float* b2  = (const float*)d_in[7];

    char* ws = (char*)d_ws;
    unsigned* gHistX = (unsigned*)(ws);                       // 48*256 u32 = 49152 B
    unsigned* gHistD = (unsigned*)(ws + 49152);               // 49152 B
    float*    partD2 = (float*)(ws + 98304);                  // 8192 f32 = 32768 B
    float*    partX2 = (float*)(ws + 98304 + 32768);          // 32768 B

    // Histograms accumulate with atomics -> zero them every call (partials are overwritten).
    hipMemsetAsync(ws, 0, 98304, stream);

    codec_main<<<B_IMG * HH, 256, 0, stream>>>(x, w0T, w0L, b0L, w1, b1, w2, b2,
                                               gHistX, gHistD, partD2, partX2);
    codec_final<<<1, 256, 0, stream>>>(gHistX, gHistD, partD2, partX2, (float*)d_out);
}